// HyperbolicDistanceHead_84799834292371
// MI455X (gfx1250) — compile-verified
//
#include <hip/hip_runtime.h>

// HyperbolicDistanceHead for MI455X (gfx1250, wave32).
// out[i,j] = -(2/sqrt(c)) * atanh( clamp( sqrt(c)*||mobius_add(-x_i, p_j)||, 0, 1-1e-5 ) )
// GEMM core: V_WMMA_F32_16X16X4_F32 (native fp32 matrix op on CDNA5), software-
// pipelined so each K-step's operand loads overlap the previous step's WMMAs.

typedef float v2f __attribute__((ext_vector_type(2)));
typedef float v8f __attribute__((ext_vector_type(8)));

#define WMMA_F32(a, b, c) \
  __builtin_amdgcn_wmma_f32_16x16x4_f32(false, (a), false, (b), (short)0, (c), false, false)

// ---------------------------------------------------------------------------
// Kernel 1: squared L2 norm of each row (one wave32 per row).
// ---------------------------------------------------------------------------
__global__ __launch_bounds__(256) void row_norm2_kernel(const float* __restrict__ src,
                                                        float* __restrict__ out,
                                                        int nrows, int D) {
  const int wave = threadIdx.x >> 5;
  const int lane = threadIdx.x & 31;
  const int row  = blockIdx.x * 8 + wave;
  if (row >= nrows) return;

  const float4* pr = (const float4*)(src + (size_t)row * (size_t)D);
  const int n4 = D >> 2;
  float s = 0.0f;
  for (int i = lane; i < n4; i += 32) {
    float4 v = pr[i];
    s += v.x * v.x + v.y * v.y + v.z * v.z + v.w * v.w;
  }
  // wave32 butterfly reduction
  for (int off = 16; off; off >>= 1) s += __shfl_xor(s, off, 32);
  if (lane == 0) out[row] = s;
}

// ---------------------------------------------------------------------------
// Kernel 2: fused WMMA f32 GEMM + Poincare-distance epilogue.
// Per wave: 16-row x 64-col output tile (4 v8f accumulators).
// Block: 256 threads = 8 waves arranged 2 (rows) x 4 (cols) -> 32 x 256 tile.
// ---------------------------------------------------------------------------
__global__ __launch_bounds__(256) void hyp_dist_kernel(const float* __restrict__ x,
                                                       const float* __restrict__ p,
                                                       const float* __restrict__ x2,
                                                       const float* __restrict__ p2,
                                                       float* __restrict__ out,
                                                       int C, int D) {
  const int lane = threadIdx.x & 31;
  const int wave = threadIdx.x >> 5;
  const int wr = wave >> 2;  // 0..1
  const int wc = wave & 3;   // 0..3

  const int row_base = blockIdx.y * 32 + wr * 16;
  const int col_base = blockIdx.x * 256 + wc * 64;

  // WMMA f32 16x16x4 operand layout (wave32):
  //   A: lane l, vgpr j -> M = l&15, K = 2*(l>>4) + j   (float2 per lane per K-step)
  //   B: lane l, vgpr j -> N = l&15, K = 2*(l>>4) + j   (symmetric; p is row-major over N)
  const int m  = lane & 15;
  const int kh = (lane >> 4) << 1;  // 0 or 2

  const float* xr  = x + (size_t)(row_base + m) * (size_t)D + kh;
  const float* pr0 = p + (size_t)(col_base + m) * (size_t)D + kh;
  const float* pr1 = pr0 + (size_t)16 * (size_t)D;
  const float* pr2 = pr0 + (size_t)32 * (size_t)D;
  const float* pr3 = pr0 + (size_t)48 * (size_t)D;

  v8f acc0 = {0, 0, 0, 0, 0, 0, 0, 0};
  v8f acc1 = acc0, acc2 = acc0, acc3 = acc0;

  // Software pipeline: operands for step k+1 are loaded before the WMMAs of
  // step k are issued, so the loadcnt wait overlaps 4 matrix ops of work.
  v2f a  = *(const v2f*)(xr);
  v2f b0 = *(const v2f*)(pr0);
  v2f b1 = *(const v2f*)(pr1);
  v2f b2 = *(const v2f*)(pr2);
  v2f b3 = *(const v2f*)(pr3);

#pragma unroll 8
  for (int kk = 4; kk < D; kk += 4) {
    v2f an  = *(const v2f*)(xr + kk);
    v2f b0n = *(const v2f*)(pr0 + kk);
    v2f b1n = *(const v2f*)(pr1 + kk);
    v2f b2n = *(const v2f*)(pr2 + kk);
    v2f b3n = *(const v2f*)(pr3 + kk);
    acc0 = WMMA_F32(a, b0, acc0);
    acc1 = WMMA_F32(a, b1, acc1);
    acc2 = WMMA_F32(a, b2, acc2);
    acc3 = WMMA_F32(a, b3, acc3);
    a = an; b0 = b0n; b1 = b1n; b2 = b2n; b3 = b3n;
  }
  // drain: last K-step
  acc0 = WMMA_F32(a, b0, acc0);
  acc1 = WMMA_F32(a, b1, acc1);
  acc2 = WMMA_F32(a, b2, acc2);
  acc3 = WMMA_F32(a, b3, acc3);

  // C/D layout: lane l, vgpr v -> M = v + 8*(l>>4), N = l&15
  const int rbase = row_base + ((lane >> 4) << 3);
  float xx[8];
#pragma unroll
  for (int v = 0; v < 8; ++v) xx[v] = x2[rbase + v];

  const float cc = 0.5f;                     // curvature c
  const float sc = 0.7071067811865476f;      // sqrt(c)
  const float inv2sc = 2.8284271247461903f;  // 2/sqrt(c)

#define EPILOGUE(ACC, T)                                                      \
  do {                                                                        \
    const int jcol = col_base + 16 * (T) + m;                                 \
    const float pp = p2[jcol];                                                \
    _Pragma("unroll") for (int v = 0; v < 8; ++v) {                           \
      const float md  = -(ACC)[v]; /* <-x, p> */                              \
      const float Af  = 1.0f + 2.0f * cc * md + cc * pp;                      \
      const float Bf  = 1.0f - cc * xx[v];                                    \
      const float den = 1.0f + 2.0f * cc * md + cc * cc * xx[v] * pp;         \
      const float num2 = Af * Af * xx[v] + 2.0f * Af * Bf * md + Bf * Bf * pp;\
      const float m2  = fmaxf(num2, 0.0f) / fmaxf(den * den, 1e-15f);         \
      float arg = sc * sqrtf(m2 + 1e-15f);                                    \
      arg = fminf(arg, 1.0f - 1e-5f);                                         \
      const float dist = inv2sc * 0.5f * __logf((1.0f + arg) / (1.0f - arg)); \
      out[(size_t)(rbase + v) * (size_t)C + jcol] = -dist;                    \
    }                                                                         \
  } while (0)

  EPILOGUE(acc0, 0);
  EPILOGUE(acc1, 1);
  EPILOGUE(acc2, 2);
  EPILOGUE(acc3, 3);
#undef EPILOGUE
}

// ---------------------------------------------------------------------------
// Host launcher
// ---------------------------------------------------------------------------
extern "C" void kernel_launch(void* const* d_in, const int* in_sizes, int n_in,
                              void* d_out, int out_size, void* d_ws, size_t ws_size,
                              hipStream_t stream) {
  const float* x = (const float*)d_in[0];      // [B, D] fp32
  const float* p = (const float*)d_in[1];      // [C, D] fp32

  const int D = 1024;
  const int B = in_sizes[0] / D;               // 128
  const int C = in_sizes[1] / D;               // 4096

  float* x2 = (float*)d_ws;                    // [B]
  float* p2 = x2 + B;                          // [C]

  row_norm2_kernel<<<(B + 7) / 8, 256, 0, stream>>>(x, x2, B, D);
  row_norm2_kernel<<<(C + 7) / 8, 256, 0, stream>>>(p, p2, C, D);

  dim3 grid(C / 256, B / 32);                  // (16, 4)
  hyp_dist_kernel<<<grid, 256, 0, stream>>>(x, p, x2, p2, (float*)d_out, C, D);
}